// CompositionalNN_17308718202922
// MI455X (gfx1250) — compile-verified
//
#include <hip/hip_runtime.h>
#include <math.h>

// Problem constants (from reference): H = 2048, S = 64.
#define HDIM 2048
#define SEQ  64

typedef __attribute__((ext_vector_type(2))) float v2f;
typedef __attribute__((ext_vector_type(8))) float v8f;

// ---------------------------------------------------------------------------
// Kernel 1: one fused LSTM cell step (both directions via blockIdx.y).
// Each block computes 16 rows m in [m_base, m_base+16) of h for one direction:
//   g = Wih @ x_last + Whh @ h0 + bih + bhh          (4 gates, same 16 rows)
//   c = sigm(g_f)*c0 + sigm(g_i)*tanh(g_g);  h = sigm(g_o)*tanh(c)
// 8 waves: wave w -> gate (w&3), K-half (w>>2). GEMV done with
// v_wmma_f32_16x16x4_f32: A = 16x4 weight tile, B = activation broadcast.
// ---------------------------------------------------------------------------
__global__ __launch_bounds__(256) void lstm_step_kernel(
    const float* __restrict__ x_last,
    const float* __restrict__ Wih_f, const float* __restrict__ Whh_f,
    const float* __restrict__ bih_f, const float* __restrict__ bhh_f,
    const float* __restrict__ h0_f,  const float* __restrict__ c0_f,
    const float* __restrict__ Wih_r, const float* __restrict__ Whh_r,
    const float* __restrict__ bih_r, const float* __restrict__ bhh_r,
    const float* __restrict__ h0_r,  const float* __restrict__ c0_r,
    float* __restrict__ h_out)   // [0,H): fwd   [H,2H): rev
{
    __shared__ float s_x[HDIM];
    __shared__ float s_h[HDIM];
    __shared__ float s_part[2][4][16];   // [k-half][gate][row]

    const int dir = blockIdx.y;
    const float* __restrict__ Wih = dir ? Wih_r : Wih_f;
    const float* __restrict__ Whh = dir ? Whh_r : Whh_f;
    const float* __restrict__ bih = dir ? bih_r : bih_f;
    const float* __restrict__ bhh = dir ? bhh_r : bhh_f;
    const float* __restrict__ h0  = dir ? h0_r  : h0_f;
    const float* __restrict__ c0  = dir ? c0_r  : c0_f;

    const int tid = threadIdx.x;

    // Preload activation vectors into LDS (broadcast operands for WMMA B).
    for (int i = tid; i < HDIM; i += 256) {
        s_x[i] = x_last[i];
        s_h[i] = h0[i];
    }
    __syncthreads();

    const int wave = tid >> 5;
    const int lane = tid & 31;
    const int gate = wave & 3;       // 0:i 1:f 2:g 3:o
    const int half = wave >> 2;      // K-half
    const int row  = lane & 15;      // M within tile (A layout)
    const int koff = (lane < 16) ? 0 : 2;   // K sub-offset for this lane half

    const int m_base = blockIdx.x * 16;
    const size_t grow = (size_t)(gate * HDIM + m_base + row) * (size_t)HDIM;
    const float* __restrict__ Wx = Wih + grow;
    const float* __restrict__ Wh = Whh + grow;

    const int k0 = half * (HDIM / 2);
    const int k1 = k0 + (HDIM / 2);

    // Two independent accumulator chains to relax WMMA RAW dependency.
    v8f accx = {};
    v8f acch = {};
    for (int k = k0; k < k1; k += 4) {
        if ((k & 31) == 0) {   // uniform: prefetch weight stream ~2KB ahead
            __builtin_prefetch(Wx + k + 512, 0, 1);
            __builtin_prefetch(Wh + k + 512, 0, 1);
        }
        v2f a0 = *(const v2f*)(Wx + k + koff);
        v2f b0 = { s_x[k + koff], s_x[k + koff + 1] };
        accx = __builtin_amdgcn_wmma_f32_16x16x4_f32(
            false, a0, false, b0, (short)0, accx, false, false);

        v2f a1 = *(const v2f*)(Wh + k + koff);
        v2f b1 = { s_h[k + koff], s_h[k + koff + 1] };
        acch = __builtin_amdgcn_wmma_f32_16x16x4_f32(
            false, a1, false, b1, (short)0, acch, false, false);
    }
    v8f acc = accx + acch;

    // D layout: lane 0 v[r] = (M=r, N=0); lane 16 v[r] = (M=8+r, N=0).
    if (lane == 0) {
        #pragma unroll
        for (int r = 0; r < 8; ++r) s_part[half][gate][r] = acc[r];
    }
    if (lane == 16) {
        #pragma unroll
        for (int r = 0; r < 8; ++r) s_part[half][gate][8 + r] = acc[r];
    }
    __syncthreads();

    if (tid < 16) {
        const int m = m_base + tid;
        float gi = s_part[0][0][tid] + s_part[1][0][tid] + bih[0 * HDIM + m] + bhh[0 * HDIM + m];
        float gf = s_part[0][1][tid] + s_part[1][1][tid] + bih[1 * HDIM + m] + bhh[1 * HDIM + m];
        float gg = s_part[0][2][tid] + s_part[1][2][tid] + bih[2 * HDIM + m] + bhh[2 * HDIM + m];
        float go = s_part[0][3][tid] + s_part[1][3][tid] + bih[3 * HDIM + m] + bhh[3 * HDIM + m];

        float ig = 1.0f / (1.0f + __expf(-gi));
        float fg = 1.0f / (1.0f + __expf(-gf));
        float cg = tanhf(gg);
        float og = 1.0f / (1.0f + __expf(-go));

        float c = fg * c0[m] + ig * cg;
        h_out[dir * HDIM + m] = og * tanhf(c);
    }
}

// ---------------------------------------------------------------------------
// Kernel 2: out = tanh(W_lin @ concat(h_f, h_r) + b_lin)
// W_lin is (H, 2H) row-major. Same WMMA-GEMV pattern; K=4096 split over 8 waves.
// ---------------------------------------------------------------------------
__global__ __launch_bounds__(256) void linear_tanh_kernel(
    const float* __restrict__ hcat,   // 2H floats (workspace)
    const float* __restrict__ W,      // H x 2H
    const float* __restrict__ bias,   // H
    float* __restrict__ out)          // H
{
    __shared__ float s_h[2 * HDIM];
    __shared__ float s_part[8][16];

    const int tid = threadIdx.x;
    for (int i = tid; i < 2 * HDIM; i += 256) s_h[i] = hcat[i];
    __syncthreads();

    const int wave = tid >> 5;
    const int lane = tid & 31;
    const int row  = lane & 15;
    const int koff = (lane < 16) ? 0 : 2;

    const int m_base = blockIdx.x * 16;
    const float* __restrict__ Wr = W + (size_t)(m_base + row) * (size_t)(2 * HDIM);

    const int kchunk = (2 * HDIM) / 8;   // 512
    const int k0 = wave * kchunk;
    const int k1 = k0 + kchunk;

    v8f acc = {};
    for (int k = k0; k < k1; k += 4) {
        if ((k & 31) == 0) {
            __builtin_prefetch(Wr + k + 512, 0, 1);
        }
        v2f a = *(const v2f*)(Wr + k + koff);
        v2f b = { s_h[k + koff], s_h[k + koff + 1] };
        acc = __builtin_amdgcn_wmma_f32_16x16x4_f32(
            false, a, false, b, (short)0, acc, false, false);
    }

    if (lane == 0) {
        #pragma unroll
        for (int r = 0; r < 8; ++r) s_part[wave][r] = acc[r];
    }
    if (lane == 16) {
        #pragma unroll
        for (int r = 0; r < 8; ++r) s_part[wave][8 + r] = acc[r];
    }
    __syncthreads();

    if (tid < 16) {
        const int m = m_base + tid;
        float s = bias[m];
        #pragma unroll
        for (int w = 0; w < 8; ++w) s += s_part[w][tid];
        out[m] = tanhf(s);
    }
}

extern "C" void kernel_launch(void* const* d_in, const int* in_sizes, int n_in,
                              void* d_out, int out_size, void* d_ws, size_t ws_size,
                              hipStream_t stream) {
    const float* x      = (const float*)d_in[0];   // (S,1,H)
    const float* h0_fwd = (const float*)d_in[1];
    const float* c0_fwd = (const float*)d_in[2];
    const float* h0_rev = (const float*)d_in[3];
    const float* c0_rev = (const float*)d_in[4];
    const float* Wih_f  = (const float*)d_in[5];
    const float* Whh_f  = (const float*)d_in[6];
    const float* bih_f  = (const float*)d_in[7];
    const float* bhh_f  = (const float*)d_in[8];
    const float* Wih_r  = (const float*)d_in[9];
    const float* Whh_r  = (const float*)d_in[10];
    const float* bih_r  = (const float*)d_in[11];
    const float* bhh_r  = (const float*)d_in[12];
    const float* W_lin  = (const float*)d_in[13];
    const float* b_lin  = (const float*)d_in[14];

    // Reference uses only scan step 0, whose input is x[-1] for BOTH directions.
    const float* x_last = x + (size_t)(SEQ - 1) * HDIM;

    float* hbuf = (float*)d_ws;   // 2*H floats: [0,H)=h_fwd, [H,2H)=h_rev

    dim3 grid1(HDIM / 16, 2);
    lstm_step_kernel<<<grid1, 256, 0, stream>>>(
        x_last,
        Wih_f, Whh_f, bih_f, bhh_f, h0_fwd, c0_fwd,
        Wih_r, Whh_r, bih_r, bhh_r, h0_rev, c0_rev,
        hbuf);

    linear_tanh_kernel<<<HDIM / 16, 256, 0, stream>>>(
        hbuf, W_lin, b_lin, (float*)d_out);
}